// KMeans_10874857193694
// MI455X (gfx1250) — compile-verified
//
#include <hip/hip_runtime.h>

#define N_POINTS   262144
#define DIM        64
#define K_CLUSTERS 256
#define MAX_ITERS  10
#define TOL        1e-4f

typedef __attribute__((ext_vector_type(2))) float v2f;
typedef __attribute__((ext_vector_type(4))) float v4f;
typedef __attribute__((ext_vector_type(8))) float v8f;

// ---------------------------------------------------------------------------
// Kernel 1: gather initial centroids, clear done flag.
// ---------------------------------------------------------------------------
__global__ __launch_bounds__(256)
void km_init(const float* __restrict__ data,
             const long long* __restrict__ init_idx,
             float* __restrict__ centroids,
             int* __restrict__ done) {
    int k = threadIdx.x;                 // 256 threads, one cluster each
    long long src = init_idx[k];
    const float* srow = data + (size_t)src * DIM;
    float* drow = centroids + (size_t)k * DIM;
#pragma unroll 8
    for (int d = 0; d < DIM; ++d) drow[d] = srow[d];
    if (k == 0) *done = 0;
}

// ---------------------------------------------------------------------------
// Kernel 2 (per iteration): zero accumulators, compute -0.5*||c_k||^2.
// ---------------------------------------------------------------------------
__global__ __launch_bounds__(256)
void km_prep(const float* __restrict__ centroids,
             float* __restrict__ c2neg,
             float* __restrict__ sums,
             float* __restrict__ counts) {
    int t = threadIdx.x;
    for (int i = t; i < K_CLUSTERS * DIM; i += blockDim.x) sums[i] = 0.f;
    counts[t] = 0.f;
    float s = 0.f;
    const float* c = centroids + (size_t)t * DIM;
#pragma unroll 8
    for (int d = 0; d < DIM; ++d) s += c[d] * c[d];
    c2neg[t] = -0.5f * s;
}

// ---------------------------------------------------------------------------
// Kernel 3 (per iteration): WMMA distance GEMM + argmin + LDS accumulation.
// One wave32 handles 16 points vs all 256 clusters.
//   score(p,k) = x_p . c_k - 0.5*||c_k||^2  (argmax == argmin of squared dist)
// Centroids live in LDS (shared by all 8 waves); B tiles preloaded to regs so
// the 16-deep WMMA chain issues without memory stalls.
// ---------------------------------------------------------------------------
__global__ __launch_bounds__(256)
void km_assign(const float* __restrict__ data,
               const float* __restrict__ centroids,
               const float* __restrict__ c2neg,
               const int* __restrict__ done,
               int* __restrict__ labels,
               float* __restrict__ gsums,
               float* __restrict__ gcounts) {
    if (*done) return;                        // wave-uniform: EXEC stays all-1s

    __shared__ float cent[K_CLUSTERS * DIM];  // 64 KB centroid cache
    __shared__ float lsum[K_CLUSTERS * DIM];  // 64 KB block-local partial sums
    __shared__ float lcnt[K_CLUSTERS];        // block-local counts
    __shared__ float lc2[K_CLUSTERS];         // -0.5*||c||^2 cache

    {   // cooperative fill: b128 copies of centroids, b128 zeroing of lsum
        const v4f* csrc = (const v4f*)centroids;
        v4f* cdst = (v4f*)cent;
        v4f* zdst = (v4f*)lsum;
        const v4f zero4 = { 0.f, 0.f, 0.f, 0.f };
        for (int i = threadIdx.x; i < K_CLUSTERS * DIM / 4; i += blockDim.x) {
            cdst[i] = csrc[i];
            zdst[i] = zero4;
        }
        lc2[threadIdx.x]  = c2neg[threadIdx.x];
        lcnt[threadIdx.x] = 0.f;
    }
    __syncthreads();

    const int lane  = threadIdx.x & 31;
    const int wave  = threadIdx.x >> 5;
    const int wpb   = blockDim.x >> 5;
    const int gwave = blockIdx.x * wpb + wave;
    const int nwav  = gridDim.x * wpb;

    const int row   = lane & 15;   // A: M row / B: N col within tile
    const int half  = lane >> 4;   // 0: K=0,1  1: K=2,3 (per 16x4 fp32 layout)
    const int khalf = half * 2;
    const int ntiles = N_POINTS / 16;

    for (int tile = gwave; tile < ntiles; tile += nwav) {
        const int base = tile * 16;

        // ---- load A (16 points x 64 dims) once: 16 k-steps of float2/lane
        v2f a[16];
        const float* xrow = data + (size_t)(base + row) * DIM + khalf;
#pragma unroll
        for (int s = 0; s < 16; ++s) a[s] = *(const v2f*)(xrow + 4 * s);

        float best[8];
        int   bidx[8];
#pragma unroll
        for (int r = 0; r < 8; ++r) { best[r] = -3.402823466e38f; bidx[r] = 0; }

        // ---- 16 cluster tiles of 16, ascending k => strict '>' keeps lowest k
#pragma unroll 1
        for (int nt = 0; nt < 16; ++nt) {
            const int col = nt * 16 + row;
            const float cn = lc2[col];
            v8f acc = { cn, cn, cn, cn, cn, cn, cn, cn };

            // preload the whole B tile from LDS so ds-loads clause & overlap
            const float* crow = cent + col * DIM + khalf;
            v2f b[16];
#pragma unroll
            for (int s = 0; s < 16; ++s) b[s] = *(const v2f*)(crow + 4 * s);

            // 16 back-to-back WMMAs, register-only dependencies
#pragma unroll
            for (int s = 0; s < 16; ++s) {
                acc = __builtin_amdgcn_wmma_f32_16x16x4_f32(
                    false, a[s], false, b[s], (short)0, acc, false, false);
            }
#pragma unroll
            for (int r = 0; r < 8; ++r) {
                if (acc[r] > best[r]) { best[r] = acc[r]; bidx[r] = col; }
            }
        }

        // ---- cross-lane argmax over the 16 columns held by each lane half
#pragma unroll
        for (int r = 0; r < 8; ++r) {
#pragma unroll
            for (int m = 1; m < 16; m <<= 1) {
                float s2 = __shfl_xor(best[r], m, 32);
                int   i2 = __shfl_xor(bidx[r], m, 32);
                if (s2 > best[r] || (s2 == best[r] && i2 < bidx[r])) {
                    best[r] = s2; bidx[r] = i2;
                }
            }
        }

        // lanes with (lane&15)<8 write labels for points base + r + 8*half
        int myIdx = bidx[0];
#pragma unroll
        for (int r = 1; r < 8; ++r) myIdx = (row == r) ? bidx[r] : myIdx;
        if (row < 8) labels[base + row + half * 8] = myIdx;

        // ---- accumulate sums/counts: 2 dims per lane per point, LDS atomics
#pragma unroll
        for (int m = 0; m < 16; ++m) {
            int lab = __shfl(bidx[m & 7], (m >> 3) << 4, 32);
            const float* prow = data + (size_t)(base + m) * DIM;
            v2f v = *(const v2f*)(prow + lane * 2);
            atomicAdd(&lsum[lab * DIM + lane * 2],     v.x);
            atomicAdd(&lsum[lab * DIM + lane * 2 + 1], v.y);
            if (lane == 0) atomicAdd(&lcnt[lab], 1.0f);
        }
    }

    __syncthreads();
    for (int i = threadIdx.x; i < K_CLUSTERS * DIM; i += blockDim.x)
        atomicAdd(&gsums[i], lsum[i]);
    atomicAdd(&gcounts[threadIdx.x], lcnt[threadIdx.x]);
}

// ---------------------------------------------------------------------------
// Kernel 4 (per iteration): centroids = sums/counts, delta check, done flag.
// ---------------------------------------------------------------------------
__global__ __launch_bounds__(256)
void km_update(float* __restrict__ centroids,
               const float* __restrict__ gsums,
               const float* __restrict__ gcounts,
               int* __restrict__ done) {
    if (*done) return;
    __shared__ float red[256];
    int k = threadIdx.x;
    float cnt = gcounts[k];
    float sq = 0.f;
    float* crow = centroids + (size_t)k * DIM;
    const float* srow = gsums + (size_t)k * DIM;
#pragma unroll 8
    for (int d = 0; d < DIM; ++d) {
        float nc = srow[d] / cnt;       // empty cluster -> 0/0 = NaN (matches ref)
        float df = nc - crow[d];
        sq += df * df;
        crow[d] = nc;
    }
    red[k] = sq;
    __syncthreads();
    for (int off = 128; off > 0; off >>= 1) {
        if (k < off) red[k] += red[k + off];
        __syncthreads();
    }
    if (k == 0 && sqrtf(red[0]) < TOL) *done = 1;
}

// ---------------------------------------------------------------------------
extern "C" void kernel_launch(void* const* d_in, const int* in_sizes, int n_in,
                              void* d_out, int out_size, void* d_ws, size_t ws_size,
                              hipStream_t stream) {
    const float*     data     = (const float*)d_in[0];
    const long long* init_idx = (const long long*)d_in[1];  // int64 indices
    int*             labels   = (int*)d_out;                // int32 labels

    float* centroids = (float*)d_ws;                          // 16384 f32
    float* sums      = centroids + K_CLUSTERS * DIM;          // 16384 f32
    float* counts    = sums      + K_CLUSTERS * DIM;          // 256 f32
    float* c2neg     = counts    + K_CLUSTERS;                // 256 f32
    int*   done      = (int*)(c2neg + K_CLUSTERS);            // 1 int

    km_init<<<1, 256, 0, stream>>>(data, init_idx, centroids, done);
    for (int it = 0; it < MAX_ITERS; ++it) {
        km_prep<<<1, 256, 0, stream>>>(centroids, c2neg, sums, counts);
        km_assign<<<256, 256, 0, stream>>>(data, centroids, c2neg, done,
                                           labels, sums, counts);
        km_update<<<1, 256, 0, stream>>>(centroids, sums, counts, done);
    }
}